// DCNNv2_63917703299186
// MI455X (gfx1250) — compile-verified
//
#include <hip/hip_runtime.h>

typedef __attribute__((ext_vector_type(2))) float v2f;
typedef __attribute__((ext_vector_type(8))) float v8f;

constexpr int N_NODES = 500000;   // divisible by 16 -> full WMMA tiles, EXEC all-ones
constexpr int N_EDGES = 8000000;
constexpr int K_EXT   = 64;
constexpr int E_EXT   = 512;

// Padded row stride (floats) for xM / msg: 16B rows -> single b128 gather per edge
constexpr int RS = 4;

// ---------------- zero msg + accumulators ----------------
__global__ void zero_kernel(float* __restrict__ p, int n) {
    int stride = gridDim.x * blockDim.x;
    for (int i = blockIdx.x * blockDim.x + threadIdx.x; i < n; i += stride)
        p[i] = 0.0f;
}

// ---------------- xM = x @ M via V_WMMA_F32_16X16X4_F32 ----------------
// One wave handles a 16-node tile. A: 16x4 (K=3 padded), B: 4x16 (3x3 M padded),
// D: 16x16 f32 (only columns 0..2 are meaningful). Output rows padded to RS floats.
__global__ void xm_kernel(const float* __restrict__ x, const float* __restrict__ Mm,
                          float* __restrict__ xM) {
    int g    = blockIdx.y;
    int wave = threadIdx.x >> 5;
    int lane = threadIdx.x & 31;
    int tile = blockIdx.x * (blockDim.x >> 5) + wave;
    int base = tile * 16;
    if (base >= N_NODES) return;                     // uniform per wave
    const float* xg = x  + (size_t)g * N_NODES * 3;
    float*       yg = xM + (size_t)g * N_NODES * RS;

    int  n  = lane & 15;
    bool hi = lane >= 16;

    // A-matrix: lane = row, vgpr0 = K{0|2}, vgpr1 = K{1|3}
    int row = base + n;
    v2f a;
    if (!hi) { a.x = xg[row * 3 + 0]; a.y = xg[row * 3 + 1]; }
    else     { a.x = xg[row * 3 + 2]; a.y = 0.0f; }          // K=3 pad

    // B-matrix: lane = col n, vgpr0 = K{0|2}, vgpr1 = K{1|3}; zero-pad n>=3, K=3
    int k0 = hi ? 2 : 0;
    v2f b;
    b.x = (n < 3)        ? Mm[k0 * 3 + n] : 0.0f;
    b.y = (n < 3 && !hi) ? Mm[1  * 3 + n] : 0.0f;

    v8f c = {};
    c = __builtin_amdgcn_wmma_f32_16x16x4_f32(false, a, false, b, (short)0, c,
                                              false, false);

    // D layout: vgpr i -> row base+i (+8 for hi half), lane&15 -> col
    if (n < 3) {
        int mbase = base + (hi ? 8 : 0);
#pragma unroll
        for (int i = 0; i < 8; ++i)
            yg[(size_t)(mbase + i) * RS + n] = c[i];
    }
}

// ---------------- edge scatter: msg[dst] += xM[src] ----------------
// 16B-aligned padded rows: one global_load_b128 gather + 3 global_atomic_add_f32
__global__ void edge_kernel(const int* __restrict__ src, const int* __restrict__ dst,
                            const float4* __restrict__ xM, float* __restrict__ msg) {
    int g = blockIdx.y;
    int i = blockIdx.x * blockDim.x + threadIdx.x;
    if (i >= N_EDGES) return;
    int s = src[(size_t)g * N_EDGES + i];
    int d = dst[(size_t)g * N_EDGES + i];
    const float4* yg = xM  + (size_t)g * N_NODES;
    float*        mg = msg + (size_t)g * N_NODES * RS;
    float4 a = yg[s];                                 // b128, pad word ignored
    unsafeAtomicAdd(&mg[(size_t)d * RS + 0], a.x);
    unsafeAtomicAdd(&mg[(size_t)d * RS + 1], a.y);
    unsafeAtomicAdd(&mg[(size_t)d * RS + 2], a.z);
}

// ---------------- sum over nodes of relu(x@W + msg), via WMMA ----------------
__global__ void hsum_kernel(const float* __restrict__ x, const float* __restrict__ W,
                            const float* __restrict__ msg, float* __restrict__ int_sum) {
    __shared__ float ssum[3];
    if (threadIdx.x < 3) ssum[threadIdx.x] = 0.0f;
    __syncthreads();

    int g    = blockIdx.y;
    int wave = threadIdx.x >> 5;
    int lane = threadIdx.x & 31;
    int tile = blockIdx.x * (blockDim.x >> 5) + wave;
    int base = tile * 16;

    int  n  = lane & 15;
    bool hi = lane >= 16;
    float partial = 0.0f;

    if (base < N_NODES) {                            // uniform per wave
        const float* xg = x   + (size_t)g * N_NODES * 3;
        const float* mg = msg + (size_t)g * N_NODES * RS;

        int row = base + n;
        v2f a;
        if (!hi) { a.x = xg[row * 3 + 0]; a.y = xg[row * 3 + 1]; }
        else     { a.x = xg[row * 3 + 2]; a.y = 0.0f; }

        int k0 = hi ? 2 : 0;
        v2f b;
        b.x = (n < 3)        ? W[k0 * 3 + n] : 0.0f;
        b.y = (n < 3 && !hi) ? W[1  * 3 + n] : 0.0f;

        v8f c = {};
        c = __builtin_amdgcn_wmma_f32_16x16x4_f32(false, a, false, b, (short)0, c,
                                                  false, false);

        if (n < 3) {
            int mbase = base + (hi ? 8 : 0);
#pragma unroll
            for (int i = 0; i < 8; ++i) {
                float h = c[i] + mg[(size_t)(mbase + i) * RS + n];
                partial += (h > 0.0f) ? h : 0.0f;
            }
            atomicAdd(&ssum[n], partial);            // LDS ds_add_f32
        }
    }
    __syncthreads();
    if (threadIdx.x < 3)
        unsafeAtomicAdd(&int_sum[g * 4 + threadIdx.x], ssum[threadIdx.x]);
}

// ---------------- tiny epilogue: external graphs + MLP + softmaxes ----------------
__device__ inline void softmax3(const float* in, float* out) {
    float m = fmaxf(in[0], fmaxf(in[1], in[2]));
    float e0 = __expf(in[0] - m), e1 = __expf(in[1] - m), e2 = __expf(in[2] - m);
    float s = e0 + e1 + e2;
    out[0] = e0 / s; out[1] = e1 / s; out[2] = e2 / s;
}

__global__ void final_kernel(const float* __restrict__ ext_rep,
                             const float* __restrict__ U, const float* __restrict__ V,
                             const float* __restrict__ l1_w, const float* __restrict__ l1_b,
                             const float* __restrict__ l2_w, const float* __restrict__ l2_b,
                             const int* __restrict__ ext_src, const int* __restrict__ ext_dst,
                             const float* __restrict__ int_sum, float* __restrict__ out) {
    __shared__ float me[K_EXT * 3];
    if (threadIdx.x != 0 || blockIdx.x != 0) return;

    float ext_emb[2][3];
    for (int g = 0; g < 2; ++g) {
        const float* rg = ext_rep + g * K_EXT * 3;
        for (int i = 0; i < K_EXT * 3; ++i) me[i] = 0.0f;
        for (int e = 0; e < E_EXT; ++e) {
            int s = ext_src[g * E_EXT + e];
            int d = ext_dst[g * E_EXT + e];
            for (int c = 0; c < 3; ++c) {
                float acc = 0.0f;
                for (int k = 0; k < 3; ++k) acc += rg[s * 3 + k] * V[k * 3 + c];
                me[d * 3 + c] += acc;
            }
        }
        float h[3];
        for (int c = 0; c < 3; ++c) {
            float acc = me[(K_EXT - 1) * 3 + c];
            for (int k = 0; k < 3; ++k) acc += rg[(K_EXT - 1) * 3 + k] * U[k * 3 + c];
            h[c] = (acc > 0.0f) ? acc : 0.0f;
        }
        softmax3(h, ext_emb[g]);
    }

    // internal softmaxes -> out[1..6]
    for (int g = 0; g < 2; ++g) {
        float s[3] = { int_sum[g * 4 + 0], int_sum[g * 4 + 1], int_sum[g * 4 + 2] };
        float sm[3];
        softmax3(s, sm);
        out[1 + g * 3 + 0] = sm[0];
        out[1 + g * 3 + 1] = sm[1];
        out[1 + g * 3 + 2] = sm[2];
    }

    // third = [e1*e2, e1+e2]; v = relu(third@l1_w + b); score = relu(v@l2_w + b)
    float t[6];
    for (int c = 0; c < 3; ++c) {
        t[c]     = ext_emb[0][c] * ext_emb[1][c];
        t[3 + c] = ext_emb[0][c] + ext_emb[1][c];
    }
    float v[3];
    for (int j = 0; j < 3; ++j) {
        float acc = l1_b[j];
        for (int i = 0; i < 6; ++i) acc += t[i] * l1_w[i * 3 + j];
        v[j] = (acc > 0.0f) ? acc : 0.0f;
    }
    float sc = l2_b[0];
    for (int j = 0; j < 3; ++j) sc += v[j] * l2_w[j];
    out[0] = (sc > 0.0f) ? sc : 0.0f;
}

extern "C" void kernel_launch(void* const* d_in, const int* in_sizes, int n_in,
                              void* d_out, int out_size, void* d_ws, size_t ws_size,
                              hipStream_t stream) {
    const float* x       = (const float*)d_in[0];
    const float* ext_rep = (const float*)d_in[1];
    const float* W       = (const float*)d_in[2];
    const float* Mm      = (const float*)d_in[3];
    const float* U       = (const float*)d_in[4];
    const float* V       = (const float*)d_in[5];
    const float* l1_w    = (const float*)d_in[6];
    const float* l1_b    = (const float*)d_in[7];
    const float* l2_w    = (const float*)d_in[8];
    const float* l2_b    = (const float*)d_in[9];
    const int*   e_src   = (const int*)d_in[10];
    const int*   e_dst   = (const int*)d_in[11];
    const int*   x_src   = (const int*)d_in[12];
    const int*   x_dst   = (const int*)d_in[13];
    float*       out     = (float*)d_out;

    // workspace layout (16B-padded node rows)
    char*  wsb     = (char*)d_ws;
    float* xM      = (float*)(wsb);                     // 2*500000*4*4B = 16,000,000 B
    float* msg     = (float*)(wsb + 16000000);          // 16,000,000 B
    float* int_sum = (float*)(wsb + 32000000);          // 8 floats (g*4+c)
    // msg + int_sum are contiguous: zero 4,000,008 floats starting at msg
    zero_kernel<<<2048, 256, 0, stream>>>(msg, 2 * N_NODES * RS + 8);

    // xM = x @ M  (WMMA), 8 waves/block -> 128 nodes per block
    const int tiles  = N_NODES / 16;                    // 31250
    const int blocks = (tiles + 7) / 8;                 // 3907
    xm_kernel<<<dim3(blocks, 2), 256, 0, stream>>>(x, Mm, xM);

    // msg[dst] += xM[src] over 8M edges per graph (b128 gather + 3x f32 atomics)
    edge_kernel<<<dim3((N_EDGES + 255) / 256, 2), 256, 0, stream>>>(
        e_src, e_dst, (const float4*)xM, msg);

    // int_sum[g] = sum_nodes relu(x@W + msg)  (WMMA + LDS reduce)
    hsum_kernel<<<dim3(blocks, 2), 256, 0, stream>>>(x, W, msg, int_sum);

    // epilogue: external graphs, softmaxes, MLP -> 7 outputs
    final_kernel<<<1, 32, 0, stream>>>(ext_rep, U, V, l1_w, l1_b, l2_w, l2_b,
                                       x_src, x_dst, int_sum, out);
}